// LightCoordAtt_90391881712299
// MI455X (gfx1250) — compile-verified
//
#include <hip/hip_runtime.h>
#include <math.h>

// LightCoordAtt on MI455X (gfx1250):
//  - bandwidth-bound: ~315 MB HBM traffic floor (~14us @ 23.3 TB/s); FLOPs negligible.
//  - pooling pass (coalesced), WMMA f32 16x16x4 for the two tiny GEMM stages,
//    float4-vectorized apply pass. x (105MB) fits L2 (192MB) so 2nd read is cheap.

typedef float v2f __attribute__((ext_vector_type(2)));
typedef float v8f __attribute__((ext_vector_type(8)));

#define N_   16
#define C_   256
#define H_   80
#define W_   80
#define MIP  8
#define L_   160   // H_ + W_

__device__ __forceinline__ float sigf(float v) {
    return 1.0f / (1.0f + __expf(-v));
}

// ---------------------------------------------------------------------------
// Kernel 1: pooling.  First R threads compute row means (x_h -> y[:, :, 0:80]),
// next R threads compute column means (x_w -> y[:, :, 80:160]).
// Column path is fully coalesced; row path reads whole cachelines per lane.
// ---------------------------------------------------------------------------
__global__ void pool_kernel(const float* __restrict__ x, float* __restrict__ y_pool) {
    const int R = N_ * C_ * H_;                 // 327680 row-mean tasks
    int tid = blockIdx.x * blockDim.x + threadIdx.x;
    if (tid < R) {
        int i  = tid % H_;
        int nc = tid / H_;
        const float* p = x + ((long)nc * H_ + i) * W_;
        float s = 0.f;
#pragma unroll
        for (int j = 0; j < W_; j += 4) {
            float4 v = *reinterpret_cast<const float4*>(p + j);
            s += v.x + v.y + v.z + v.w;
        }
        y_pool[(long)nc * L_ + i] = s * (1.0f / (float)W_);
    } else {
        int t = tid - R;
        if (t < N_ * C_ * W_) {
            int j  = t % W_;
            int nc = t / W_;
            const float* p = x + (long)nc * H_ * W_ + j;
            float s = 0.f;
#pragma unroll 8
            for (int i = 0; i < H_; ++i) s += p[(long)i * W_];
            y_pool[(long)nc * L_ + H_ + j] = s * (1.0f / (float)H_);
        }
    }
}

// ---------------------------------------------------------------------------
// Kernel 2: y_mid[n,m,l] = swish(BN(sum_c w_fc[m,c] * y_pool[n,c,l]))
// One wave per (n, 16-col tile) -> grid (10, 16) = 160 waves.
// WMMA f32 16x16x4: M=16 (rows 8..15 masked to zero), K loop 256/4.
//   A layout (16x4 f32): lanes 0-15 K={k,k+1}; lanes 16-31 K={k+2,k+3}
//   B layout (4x16 f32): same K striping, N = lane&15
//   D layout: vgpr r = row r (lanes 0-15) / row r+8 (lanes 16-31)
// Two independent accumulator chains (even/odd K chunks) break the serial
// WMMA->WMMA RAW dependency; their sum equals the single-chain result.
// A-row index is clamped (l16&7) + multiplied by a 0/1 mask so loads are
// unconditional -> no exec-mask save/restore inside the WMMA loop.
// ---------------------------------------------------------------------------
__global__ void gemm1_bn_swish_kernel(const float* __restrict__ y_pool,
                                      const float* __restrict__ w_fc,
                                      const float* __restrict__ g,
                                      const float* __restrict__ b,
                                      const float* __restrict__ mean,
                                      const float* __restrict__ var,
                                      float* __restrict__ y_mid) {
    const int tile = blockIdx.x;    // 0..9
    const int n    = blockIdx.y;    // 0..15
    const int lane = threadIdx.x;
    const int half = lane >> 4;
    const int l16  = lane & 15;

    const float amask = (l16 < MIP) ? 1.0f : 0.0f;   // zero-pad rows 8..15
    const float* wrow = w_fc + (l16 & (MIP - 1)) * C_;
    const float* yb   = y_pool + (long)n * C_ * L_;
    const int    col  = tile * 16 + l16;

    v8f acc0 = {};
    v8f acc1 = {};
    for (int kk = 0; kk < C_; kk += 8) {
        {   // chain 0: K chunk [kk, kk+4)
            const int k = kk + 2 * half;
            v2f a, bb;
            a.x  = wrow[k]     * amask;
            a.y  = wrow[k + 1] * amask;
            bb.x = yb[(long)k * L_ + col];
            bb.y = yb[(long)(k + 1) * L_ + col];
            acc0 = __builtin_amdgcn_wmma_f32_16x16x4_f32(false, a, false, bb,
                                                         (short)0, acc0, false, false);
        }
        {   // chain 1: K chunk [kk+4, kk+8)
            const int k = kk + 4 + 2 * half;
            v2f a, bb;
            a.x  = wrow[k]     * amask;
            a.y  = wrow[k + 1] * amask;
            bb.x = yb[(long)k * L_ + col];
            bb.y = yb[(long)(k + 1) * L_ + col];
            acc1 = __builtin_amdgcn_wmma_f32_16x16x4_f32(false, a, false, bb,
                                                         (short)0, acc1, false, false);
        }
    }

    if (half == 0) {   // rows 0..7 hold the real (mip=8) result
#pragma unroll
        for (int r = 0; r < MIP; ++r) {
            float iv    = g[r] * rsqrtf(var[r] + 1e-5f);
            float shift = b[r] - mean[r] * iv;
            float v = (acc0[r] + acc1[r]) * iv + shift;
            v = v * sigf(v);                       // swish
            y_mid[((long)n * MIP + r) * L_ + col] = v;
        }
    }
}

// ---------------------------------------------------------------------------
// Kernel 3: a_h[n,c,l] = sigmoid(sum_m w_h[c,m] * y_mid[n,m,l])  (cols 0..79)
//           a_w[n,c,l] = sigmoid(sum_m w_w[c,m] * y_mid[n,m,l])  (cols 80..159)
// Grid (ntile=10, mtile=16, n=16), one wave each, 2 WMMAs (K=8).
// ---------------------------------------------------------------------------
__global__ void gemm2_sig_kernel(const float* __restrict__ y_mid,
                                 const float* __restrict__ w_h,
                                 const float* __restrict__ w_w,
                                 float* __restrict__ a_h,
                                 float* __restrict__ a_w) {
    const int ntile = blockIdx.x;   // 0..9  (cols)
    const int mtile = blockIdx.y;   // 0..15 (channel tiles)
    const int n     = blockIdx.z;
    const int lane  = threadIdx.x;
    const int half  = lane >> 4;
    const int l16   = lane & 15;

    const bool  is_h = (ntile < 5);              // 80 = 5 tiles of 16
    const float* Wm  = is_h ? w_h : w_w;
    const int   col  = ntile * 16 + l16;
    const int   chA  = mtile * 16 + l16;
    const float* ymb = y_mid + (long)n * MIP * L_;

    v8f acc = {};
#pragma unroll
    for (int kk = 0; kk < MIP; kk += 4) {
        const int k = kk + 2 * half;
        v2f a, bb;
        a.x  = Wm[chA * MIP + k];
        a.y  = Wm[chA * MIP + k + 1];
        bb.x = ymb[(long)k * L_ + col];
        bb.y = ymb[(long)(k + 1) * L_ + col];
        acc = __builtin_amdgcn_wmma_f32_16x16x4_f32(false, a, false, bb,
                                                    (short)0, acc, false, false);
    }

    float* dst = is_h ? a_h : a_w;
    const int cc = is_h ? col : (col - H_);
#pragma unroll
    for (int r = 0; r < MIP; ++r) {
        const int ch = mtile * 16 + half * 8 + r;
        dst[((long)n * C_ + ch) * H_ + cc] = sigf(acc[r]);
    }
}

// ---------------------------------------------------------------------------
// Kernel 4: out[n,c,i,j] = x[n,c,i,j] * a_h[n,c,i] * a_w[n,c,j]
// float4-vectorized streaming pass (the bandwidth-critical kernel).
// ---------------------------------------------------------------------------
__global__ void apply_kernel(const float* __restrict__ x,
                             const float* __restrict__ a_h,
                             const float* __restrict__ a_w,
                             float* __restrict__ out) {
    const long t = (long)blockIdx.x * blockDim.x + threadIdx.x;  // vec4 index
    const long e = t * 4;
    const int  j  = (int)(e % W_);       // 80 % 4 == 0 -> j..j+3 in one row
    const long rid = e / W_;             // (n*C + c)*H + i
    const int  i  = (int)(rid % H_);
    const long nc = rid / H_;

    const float  ah = a_h[nc * H_ + i];
    const float4 aw = *reinterpret_cast<const float4*>(a_w + nc * W_ + j);
    const float4 xv = *reinterpret_cast<const float4*>(x + e);

    float4 o;
    o.x = xv.x * ah * aw.x;
    o.y = xv.y * ah * aw.y;
    o.z = xv.z * ah * aw.z;
    o.w = xv.w * ah * aw.w;
    *reinterpret_cast<float4*>(out + e) = o;
}

// ---------------------------------------------------------------------------
// Launch
// ---------------------------------------------------------------------------
extern "C" void kernel_launch(void* const* d_in, const int* in_sizes, int n_in,
                              void* d_out, int out_size, void* d_ws, size_t ws_size,
                              hipStream_t stream) {
    const float* x     = (const float*)d_in[0];
    const float* w_fc  = (const float*)d_in[1];
    const float* bn_g  = (const float*)d_in[2];
    const float* bn_b  = (const float*)d_in[3];
    const float* bn_m  = (const float*)d_in[4];
    const float* bn_v  = (const float*)d_in[5];
    const float* w_h   = (const float*)d_in[6];
    const float* w_w   = (const float*)d_in[7];
    float* out = (float*)d_out;

    // Workspace layout (floats):
    float* ws     = (float*)d_ws;
    float* y_pool = ws;                                   // 16*256*160 = 655360
    float* y_mid  = y_pool + (long)N_ * C_ * L_;          // 16*8*160   = 20480
    float* a_h    = y_mid  + (long)N_ * MIP * L_;         // 16*256*80  = 327680
    float* a_w    = a_h    + (long)N_ * C_ * H_;          // 16*256*80  = 327680
    // total ~5.3 MB

    // 1) pooling: 2*327680 tasks
    {
        int total = N_ * C_ * (H_ + W_);
        pool_kernel<<<(total + 255) / 256, 256, 0, stream>>>(x, y_pool);
    }
    // 2) squeeze GEMM + BN + swish: one wave per (col-tile, batch)
    gemm1_bn_swish_kernel<<<dim3(L_ / 16, N_), 32, 0, stream>>>(y_pool, w_fc, bn_g,
                                                                bn_b, bn_m, bn_v,
                                                                y_mid);
    // 3) excitation GEMMs + sigmoid: one wave per 16x16 output tile
    gemm2_sig_kernel<<<dim3(L_ / 16, C_ / 16, N_), 32, 0, stream>>>(y_mid, w_h, w_w,
                                                                    a_h, a_w);
    // 4) apply attention (bandwidth-critical)
    {
        long nvec4 = (long)N_ * C_ * H_ * W_ / 4;         // 6,553,600
        apply_kernel<<<(unsigned)(nvec4 / 256), 256, 0, stream>>>(x, a_h, a_w, out);
    }
}